// LSTM_56324201120461
// MI455X (gfx1250) — compile-verified
//
#include <hip/hip_runtime.h>

// LSTM: bs=64, T=512, SI=512, SH=1024. fp32 throughout (matches reference).
// Strategy: one kernel launch per time step (graph-friendly), fp32 WMMA
// (v_wmma_f32_16x16x4_f32) for the fused GEMM g = X_t@Wx + h@Wh + b.
// Weights (24 MB) stay resident in the 192 MB L2 across all 512 steps.

#define K_SI 512
#define K_SH 1024
#define K_BS 64
#define K_T  512
#define K_NG 4096  // 4*SH

typedef __attribute__((ext_vector_type(2))) float v2f;
typedef __attribute__((ext_vector_type(8))) float v8f;

__device__ __forceinline__ float sigmoidf_(float x) {
    return 1.0f / (1.0f + __expf(-x));
}

// Grid: 256 blocks = (mt in 0..3) x (jt in 0..63). Block: 128 threads = 4 waves.
// Wave w computes gate w's 16x16 tile at rows [16*mt,16*mt+16), gate-cols
// [16*jt, 16*jt+16) via WMMA f32 16x16x4 over K = SI (X@Wx) then K = SH (h@Wh).
__global__ __launch_bounds__(128)
void lstm_step_kernel(const float* __restrict__ X,
                      const float* __restrict__ Wx,
                      const float* __restrict__ Wh,
                      const float* __restrict__ bias,
                      const float* __restrict__ h_in,
                      float* __restrict__ h_out,
                      float* __restrict__ c_buf,
                      float* __restrict__ out,
                      int t)
{
    __shared__ float lds_g[4][16][17];  // [gate][row][col], padded vs bank conflicts

    const int tid  = threadIdx.x;
    const int gate = tid >> 5;          // wave id == gate id (f, i, c, o)
    const int lane = tid & 31;
    const int lo16 = lane & 15;
    const int hi   = lane >> 4;         // 0: K+0/1 half, 1: K+2/3 half

    const int jt = blockIdx.x & 63;     // 64 column tiles of 16 (j in [0,1024))
    const int mt = blockIdx.x >> 6;     // 4 row tiles of 16 (batch rows)
    const int j0 = jt << 4;
    const int m0 = mt << 4;

    const int ncol = gate * K_SH + j0 + lo16;   // column in stacked gate space [0,4096)

    v8f acc = {};

    // ---- X_t @ Wx : K = SI ----
    // A fragment: lane holds row (m0+lo16), K = k0 + 2*hi + {0,1}  -> b64 load
    const float* Xrow = X + (size_t)(m0 + lo16) * (K_T * K_SI) + (size_t)t * K_SI;
    for (int k0 = 0; k0 < K_SI; k0 += 4) {
        const int kb = k0 + 2 * hi;
        v2f a = *(const v2f*)(Xrow + kb);
        v2f bm;
        bm.x = Wx[(size_t)kb * K_NG + ncol];
        bm.y = Wx[(size_t)(kb + 1) * K_NG + ncol];
        acc = __builtin_amdgcn_wmma_f32_16x16x4_f32(
            false, a, false, bm, (short)0, acc, false, false);
    }

    // ---- h_{t-1} @ Wh : K = SH ----
    const float* Hrow = h_in + (size_t)(m0 + lo16) * K_SH;
    for (int k0 = 0; k0 < K_SH; k0 += 4) {
        const int kb = k0 + 2 * hi;
        v2f a = *(const v2f*)(Hrow + kb);
        v2f bm;
        bm.x = Wh[(size_t)kb * K_NG + ncol];
        bm.y = Wh[(size_t)(kb + 1) * K_NG + ncol];
        acc = __builtin_amdgcn_wmma_f32_16x16x4_f32(
            false, a, false, bm, (short)0, acc, false, false);
    }

    // C/D layout: lane l, vgpr jj -> M = jj + 8*(l>>4), N = l&15
    const float bv = bias[ncol];
#pragma unroll
    for (int jj = 0; jj < 8; ++jj) {
        lds_g[gate][jj + 8 * hi][lo16] = acc[jj] + bv;
    }
    __syncthreads();

    // Elementwise cell update: 16x16 = 256 elements, 128 threads -> 2 each
    for (int idx = tid; idx < 256; idx += 128) {
        const int r = idx >> 4;
        const int cc = idx & 15;
        const int b = m0 + r;
        const int j = j0 + cc;
        const float fg = sigmoidf_(lds_g[0][r][cc]);
        const float ig = sigmoidf_(lds_g[1][r][cc]);
        const float gg = tanhf(lds_g[2][r][cc]);
        const float og = sigmoidf_(lds_g[3][r][cc]);
        const size_t cj = (size_t)b * K_SH + j;
        const float cn = fg * c_buf[cj] + ig * gg;
        const float hn = og * tanhf(cn);
        c_buf[cj] = cn;
        h_out[cj] = hn;
        out[(size_t)t * (K_BS * K_SH) + cj] = hn;
        if (t == K_T - 1) {
            // final h and c appended after the sequence in d_out
            out[(size_t)K_T * K_BS * K_SH + cj] = hn;
            out[(size_t)K_T * K_BS * K_SH + (size_t)K_BS * K_SH + cj] = cn;
        }
    }
}

__global__ void lstm_init_kernel(float* __restrict__ h0, float* __restrict__ c0) {
    const int i = blockIdx.x * blockDim.x + threadIdx.x;
    if (i < K_BS * K_SH) {
        h0[i] = 0.0f;
        c0[i] = 0.0f;
    }
}

extern "C" void kernel_launch(void* const* d_in, const int* in_sizes, int n_in,
                              void* d_out, int out_size, void* d_ws, size_t ws_size,
                              hipStream_t stream) {
    const float* X  = (const float*)d_in[0];   // (64, 512, 512)
    const float* Wx = (const float*)d_in[1];   // (512, 4096)
    const float* Wh = (const float*)d_in[2];   // (1024, 4096)
    const float* b  = (const float*)d_in[3];   // (4096,)
    float* out = (float*)d_out;                // seq (512,64,1024) ++ h ++ c

    // workspace: h ping, h pong, c  (each 64*1024 f32 = 256 KB)
    float* h0 = (float*)d_ws;
    float* h1 = h0 + K_BS * K_SH;
    float* cb = h1 + K_BS * K_SH;

    lstm_init_kernel<<<(K_BS * K_SH + 255) / 256, 256, 0, stream>>>(h0, cb);

    for (int t = 0; t < K_T; ++t) {
        const float* hin = (t & 1) ? h1 : h0;
        float*       hout = (t & 1) ? h0 : h1;
        lstm_step_kernel<<<dim3(256), dim3(128), 0, stream>>>(
            X, Wx, Wh, b, hin, hout, cb, out, t);
    }
}